// HyperAttention_42752104464609
// MI455X (gfx1250) — compile-verified
//
#include <hip/hip_runtime.h>
#include <math.h>

// HyperAttention for MI455X / gfx1250 (wave32, WMMA).
// L=32 layers, K=8 client models, D=262144 params per key. All fp32.
//
// Bandwidth-bound (AI ~ 2.7 FLOP/byte): minimum DRAM traffic is delta read
// twice + last_param once + output once ~= 1 GB -> ~45 us @ 23.3 TB/s.
// V_WMMA_F32_16X16X4_F32 keeps VALU work per byte tiny, and delta stays
// regular-temporal in L2 (192 MB) between the two passes while last/out are
// marked non-temporal (read-once / write-once streams).

typedef float v2f __attribute__((ext_vector_type(2)));
typedef float v8f __attribute__((ext_vector_type(8)));

#define L_DIM 32
#define K_DIM 8
#define D_DIM 262144
#define PAIRS 16   // two layers packed per 16x16 WMMA tile (K=8 rows each)
#define NI    16   // d-tile iterations per block in apply pass

// ---------------------------------------------------------------------------
// Pass 1: partial Gram matrices G = delta_pair (16 x D) * delta_pair^T via
// V_WMMA_F32_16X16X4_F32. For f32 16x4, the A layout (lane: M=lane&15,
// K=2*(lane>>4)+vgpr) and 4x16 B layout (lane: N=lane&15, K=2*(lane>>4)+vgpr)
// are element-identical, so the SAME v2f register feeds both operands for a
// Gram product: one global b64 load per lane per WMMA step.
// Loop is scalar-controlled (uniform trip count) so EXEC stays all-1s with no
// per-lane exec bookkeeping, and unroll-4 keeps 4 b64 loads in flight.
// ---------------------------------------------------------------------------
__global__ __launch_bounds__(256) void gram_partial_kernel(
    const float* __restrict__ delta, float* __restrict__ partial, int S)
{
    const int pair  = blockIdx.x;
    const int split = blockIdx.y;
    const int tid   = threadIdx.x;
    const int wave  = tid >> 5;
    const int lane  = tid & 31;
    const int half  = lane >> 4;
    const int lmod  = lane & 15;

    const int Dsub    = D_DIM / S;
    const int perWave = Dsub >> 3;               // 8 waves per block
    const int wstart  = split * Dsub + wave * perWave;
    const int iters   = perWave >> 2;            // one WMMA consumes K=4 cols

    // 16 rows of the pair are contiguous: (pair*2)*K*D .. +16*D
    const float* p = delta + (size_t)pair * 16 * D_DIM + (size_t)lmod * D_DIM
                   + wstart + 2 * half;

    v8f acc = {};
#pragma unroll 4
    for (int it = 0; it < iters; ++it) {
        v2f a;
        a.x = p[0];
        a.y = p[1];
        acc = __builtin_amdgcn_wmma_f32_16x16x4_f32(
            false, a, false, a, (short)0, acc, false, false);
        p += 4;
    }

    // Block reduction of the 8 per-wave 16x16 partials (fixed order).
    __shared__ float red[8][256];
#pragma unroll
    for (int i = 0; i < 8; ++i) red[wave][lane * 8 + i] = acc[i];
    __syncthreads();

    float s = 0.f;
#pragma unroll
    for (int w = 0; w < 8; ++w) s += red[w][tid];
    // Flat index t = lane*8 + i  <->  M = i + 8*(lane>>4), N = lane&15
    partial[(size_t)(pair * S + split) * 256 + tid] = s;
}

// ---------------------------------------------------------------------------
// Pass 1b: reduce split partials (fixed order), scale by D^-0.5, softmax over
// the last axis. One 64-thread block per layer; thread (k,m) owns scores[k][m].
// ---------------------------------------------------------------------------
__global__ __launch_bounds__(64) void softmax_kernel(
    const float* __restrict__ partial, float* __restrict__ attn,
    int S, float scale)
{
    const int l   = blockIdx.x;          // 0..31
    const int tid = threadIdx.x;         // 0..63
    const int k = tid >> 3, m = tid & 7;
    const int pair = l >> 1;
    const int hb = (l & 1) * 8;          // which 8x8 diagonal block of the 16x16
    const int M = hb + k, N = hb + m;
    const int lane  = ((M >= 8) ? 16 : 0) + N;   // lane&15 == N
    const int flatT = lane * 8 + (M & 7);

    const size_t base = (size_t)(pair * S) * 256;
    float sc = 0.f;
    for (int s = 0; s < S; ++s) sc += partial[base + (size_t)s * 256 + flatT];
    sc *= scale;

    __shared__ float row[8][8];
    row[k][m] = sc;
    __syncthreads();

    float mx = row[k][0];
#pragma unroll
    for (int j = 1; j < 8; ++j) mx = fmaxf(mx, row[k][j]);
    float sum = 0.f;
#pragma unroll
    for (int j = 0; j < 8; ++j) sum += __expf(row[k][j] - mx);
    attn[l * 64 + k * 8 + m] = __expf(sc - mx) / sum;
}

// ---------------------------------------------------------------------------
// Pass 2: out = last + A_final * delta, where A_final (16x16, block-diagonal
// over the layer pair) = g*attn + (1-g)*I absorbs the softmax mix, the gamma
// clamp/interpolation AND the identity term -> 4 chained WMMAs produce
// new_delta directly in C/D layout; then one add of last_param per element.
// last loads / out stores are non-temporal (single-use streams); delta loads
// stay regular-temporal so pass-1 residue in the 192 MB L2 can be reused.
// ---------------------------------------------------------------------------
__global__ __launch_bounds__(256) void apply_kernel(
    const float* __restrict__ delta, const float* __restrict__ last,
    const float* __restrict__ gamma, const float* __restrict__ attn,
    float* __restrict__ out)
{
    const int pair  = blockIdx.x;
    const int chunk = blockIdx.y;
    const int tid   = threadIdx.x;
    const int wave  = tid >> 5;
    const int lane  = tid & 31;
    const int half  = lane >> 4;
    const int lmod  = lane & 15;

    // Build A tiles once per block. A-layout: lane holds M=lane&15,
    // K-index kk = 4*j + 2*half + v  (tile j, vgpr v).
    const int M      = lmod;
    const int layerM = pair * 2 + (M >> 3);
    const int kM     = M & 7;
    const float gM   = fminf(fmaxf(gamma[layerM * 8 + kM], 0.f), 1.f);
    const float* arow = attn + layerM * 64 + kM * 8;

    v2f a[4];
#pragma unroll
    for (int j = 0; j < 4; ++j) {
#pragma unroll
        for (int v = 0; v < 2; ++v) {
            const int kk = j * 4 + half * 2 + v;
            float val = ((kk >> 3) == (M >> 3)) ? gM * arow[kk & 7] : 0.f;
            if (kk == M) val += 1.f - gM;      // (1-g)*I folded into A
            a[j][v] = val;
        }
    }

    const size_t base = (size_t)pair * 16 * D_DIM;
    const float* dbase = delta + base;
    const float* lbase = last + base;
    float*       obase = out + base;

    const int colbase = chunk * (NI * 128) + wave * 16 + lmod;
    for (int it = 0; it < NI; ++it) {
        const int col = colbase + it * 128;
        v8f c = {};
#pragma unroll
        for (int j = 0; j < 4; ++j) {
            // B-layout: lane holds N=lane&15 (=col), row kk = 4*j + 2*half + v
            const int r0 = j * 4 + half * 2;
            v2f b;
            b.x = dbase[(size_t)r0 * D_DIM + col];
            b.y = dbase[(size_t)(r0 + 1) * D_DIM + col];
            c = __builtin_amdgcn_wmma_f32_16x16x4_f32(
                false, a[j], false, b, (short)0, c, false, false);
        }
        // C/D layout: vgpr i <-> row M = i + 8*half, N = lane&15 = col
#pragma unroll
        for (int i = 0; i < 8; ++i) {
            const size_t idx = (size_t)(i + 8 * half) * D_DIM + col;
            const float lp = __builtin_nontemporal_load(lbase + idx);
            __builtin_nontemporal_store(lp + c[i], obase + idx);
        }
    }
}

// ---------------------------------------------------------------------------
extern "C" void kernel_launch(void* const* d_in, const int* in_sizes, int n_in,
                              void* d_out, int out_size, void* d_ws, size_t ws_size,
                              hipStream_t stream)
{
    (void)in_sizes; (void)n_in; (void)out_size;

    const float* last  = (const float*)d_in[0];   // last_param (L,K,D)
    const float* delta = (const float*)d_in[1];   // delta      (L,K,D)
    const float* gamma = (const float*)d_in[2];   // gamma      (L,K)
    float* out = (float*)d_out;

    // Pick split count to fit workspace (deterministic for a given ws_size).
    int S = 32;
    while (S > 1 &&
           (size_t)(PAIRS * S * 256 + L_DIM * 64) * sizeof(float) > ws_size)
        S >>= 1;

    float* partial = (float*)d_ws;                       // PAIRS*S*256 floats
    float* attnbuf = partial + (size_t)PAIRS * S * 256;  // L*64 floats

    gram_partial_kernel<<<dim3(PAIRS, S), 256, 0, stream>>>(delta, partial, S);
    softmax_kernel<<<L_DIM, 64, 0, stream>>>(partial, attnbuf, S,
                                             1.0f / 512.0f);  // D^-0.5, sqrt(262144)=512
    apply_kernel<<<dim3(PAIRS, D_DIM / (NI * 128)), 256, 0, stream>>>(
        delta, last, gamma, attnbuf, out);
}